// aspect_augumentation_55224689492133
// MI455X (gfx1250) — compile-verified
//
#include <hip/hip_runtime.h>
#include <hip/hip_bf16.h>

#define LEAKY_SLOPE 0.2f

typedef __attribute__((ext_vector_type(2)))  float    v2f;
typedef __attribute__((ext_vector_type(8)))  float    v8f;

// ---------------------------------------------------------------------------
// Kernel 1: scores = softmax(leaky_relu(users @ relation_k), axis=1)
// One wave handles a 16-user tile. K=64, N padded 3 -> 16 with a branch-free
// mask (no EXEC divergence between WMMAs). V_WMMA_F32_16X16X4_F32, full fp32.
// ---------------------------------------------------------------------------
__global__ __launch_bounds__(256) void scores_wmma_kernel(
    const int* __restrict__ user_id,
    const float* __restrict__ user_factors,
    const float* __restrict__ relation_k,
    float* __restrict__ scores_out,   // (B,3) row-major
    int B)
{
    const int lane = threadIdx.x & 31;
    const int wave = threadIdx.x >> 5;
    const int tile = blockIdx.x * (blockDim.x >> 5) + wave;
    const int row0 = tile * 16;
    if (row0 >= B) return;                       // uniform per wave

    const int m    = lane & 15;                  // A row / B col handled by lane
    const int half = lane >> 4;                  // 0: lanes 0-15, 1: lanes 16-31

    int urow_idx = row0 + m;
    if (urow_idx >= B) urow_idx = B - 1;         // clamp (B is multiple of 16)
    const int uid = user_id[urow_idx];
    const float* __restrict__ urow = user_factors + (long long)uid * 64;

    // ---- Preload B (4x16 chunks for all 16 k-steps), branch-free padding ----
    // Lane holds column n=m (clamped) with K = {4i+2*half, 4i+2*half+1};
    // lanes with m>=3 read column 0 (in-bounds) and zero it via bmask.
    const float bmask = (m < 3) ? 1.0f : 0.0f;
    const int   n     = (m < 3) ? m : 0;
    float bk[32];
    #pragma unroll
    for (int i = 0; i < 16; ++i) {
        const int kk = 4 * i + 2 * half;
        bk[2 * i + 0] = relation_k[(kk + 0) * 3 + n] * bmask;
        bk[2 * i + 1] = relation_k[(kk + 1) * 3 + n] * bmask;
    }

    v8f acc = {};
    #pragma unroll
    for (int i = 0; i < 16; ++i) {
        const int kk = 4 * i + 2 * half;
        // A 16x4 f32: lane m holds K = {kk, kk+1}
        v2f a;
        a.x = urow[kk + 0];
        a.y = urow[kk + 1];
        v2f b;
        b.x = bk[2 * i + 0];
        b.y = bk[2 * i + 1];
        acc = __builtin_amdgcn_wmma_f32_16x16x4_f32(
            false, a, false, b, (short)0, acc, false, false);
    }

    // D layout: VGPR j -> row (j + 8*half), col = lane&15. Columns 0..2 live in
    // lanes {base, base+1, base+2}. Pull them with shuffles, finish softmax.
    const int base = half << 4;                  // 0 or 16
    #pragma unroll
    for (int j = 0; j < 8; ++j) {
        const float v = acc[j];
        float s0 = __shfl(v, base + 0, 32);
        float s1 = __shfl(v, base + 1, 32);
        float s2 = __shfl(v, base + 2, 32);
        s0 = (s0 >= 0.0f) ? s0 : LEAKY_SLOPE * s0;
        s1 = (s1 >= 0.0f) ? s1 : LEAKY_SLOPE * s1;
        s2 = (s2 >= 0.0f) ? s2 : LEAKY_SLOPE * s2;
        const float mx = fmaxf(s0, fmaxf(s1, s2));
        const float e0 = __expf(s0 - mx);
        const float e1 = __expf(s1 - mx);
        const float e2 = __expf(s2 - mx);
        const float inv = 1.0f / (e0 + e1 + e2);
        const int row = row0 + j + (half << 3);
        if (lane == base && row < B) {
            float* o = scores_out + (long long)row * 3;
            o[0] = e0 * inv;
            o[1] = e1 * inv;
            o[2] = e2 * inv;
        }
    }
}

// ---------------------------------------------------------------------------
// Kernel 2: segmented gather-dot-mean + final prediction.
// seg arrays are sorted runs (repeat(arange(B), len)) -> binary-search bounds.
// One wave per segment; lane l owns dims {2l, 2l+1}; wave-shuffle reduction.
// ---------------------------------------------------------------------------
__device__ __forceinline__ int lower_bound_i32(const int* __restrict__ seg,
                                               int lo, int hi, int key)
{
    while (lo < hi) {
        const int mid = (lo + hi) >> 1;
        if (seg[mid] < key) lo = mid + 1; else hi = mid;
    }
    return lo;
}

__device__ __forceinline__ float seg_mean_dot(const int* __restrict__ ids,
                                              const int* __restrict__ seg,
                                              int T, int b,
                                              float ux, float uy,
                                              const float* __restrict__ ef,
                                              int lane)
{
    const int start = lower_bound_i32(seg, 0, T, b);
    // run lengths are short (<80); cap the second search window
    int hi = start + 128; if (hi > T) hi = T;
    const int end = lower_bound_i32(seg, start, hi, b + 1);

    float acc0 = 0.0f, acc1 = 0.0f;
    int t = start;
    for (; t + 1 < end; t += 2) {
        const int id0 = ids[t];
        const int id1 = ids[t + 1];
        const float2 e0 = *(const float2*)(ef + (long long)id0 * 64 + lane * 2);
        const float2 e1 = *(const float2*)(ef + (long long)id1 * 64 + lane * 2);
        acc0 = fmaf(e0.x, ux, fmaf(e0.y, uy, acc0));
        acc1 = fmaf(e1.x, ux, fmaf(e1.y, uy, acc1));
    }
    if (t < end) {
        const int id0 = ids[t];
        const float2 e0 = *(const float2*)(ef + (long long)id0 * 64 + lane * 2);
        acc0 = fmaf(e0.x, ux, fmaf(e0.y, uy, acc0));
    }
    float acc = acc0 + acc1;
    #pragma unroll
    for (int off = 16; off > 0; off >>= 1)
        acc += __shfl_xor(acc, off, 32);
    return acc / (float)(end - start);
}

__global__ __launch_bounds__(256) void contrib_kernel(
    const int* __restrict__ user_id,
    const int* __restrict__ a_ids, const int* __restrict__ a_seg, int Ta,
    const int* __restrict__ c_ids, const int* __restrict__ c_seg, int Tc,
    const float* __restrict__ user_factors,
    const float* __restrict__ entity_factors,
    const float* __restrict__ scores,     // (B,3) from kernel 1
    float* __restrict__ pred_out,
    float* __restrict__ ca_out,
    float* __restrict__ cc_out,
    int B)
{
    const int lane = threadIdx.x & 31;
    const int b = blockIdx.x * (blockDim.x >> 5) + (threadIdx.x >> 5);
    if (b >= B) return;                          // uniform per wave

    const int uid = user_id[b];
    const float2 u = *(const float2*)(user_factors + (long long)uid * 64 + lane * 2);

    const float c_a = seg_mean_dot(a_ids, a_seg, Ta, b, u.x, u.y, entity_factors, lane);
    const float c_c = seg_mean_dot(c_ids, c_seg, Tc, b, u.x, u.y, entity_factors, lane);

    if (lane == 0) {
        ca_out[b] = c_a;
        cc_out[b] = c_c;
        const float s_a = scores[(long long)b * 3 + 0];
        const float s_c = scores[(long long)b * 3 + 1];
        pred_out[b] = (c_a * s_a + c_c * s_c) / (s_a + s_c);
    }
}

// ---------------------------------------------------------------------------
extern "C" void kernel_launch(void* const* d_in, const int* in_sizes, int n_in,
                              void* d_out, int out_size, void* d_ws, size_t ws_size,
                              hipStream_t stream)
{
    const int*   user_id = (const int*)  d_in[0];
    const int*   a_ids   = (const int*)  d_in[1];
    const int*   a_seg   = (const int*)  d_in[2];
    const int*   c_ids   = (const int*)  d_in[3];
    const int*   c_seg   = (const int*)  d_in[4];
    const float* uf      = (const float*)d_in[5];
    const float* ef      = (const float*)d_in[6];
    const float* rk      = (const float*)d_in[7];

    const int B  = in_sizes[0];
    const int Ta = in_sizes[1];
    const int Tc = in_sizes[3];

    float* out    = (float*)d_out;
    float* pred   = out;                      // (B,)
    float* scores = out + (long long)B;       // (B,3)
    float* ca     = out + 4LL * B;            // (B,)
    float* cc     = out + 5LL * B;            // (B,)

    // Kernel 1: 16 users per wave, 8 waves (256 threads) per block
    const int tiles = (B + 15) / 16;
    scores_wmma_kernel<<<(tiles + 7) / 8, 256, 0, stream>>>(user_id, uf, rk, scores, B);

    // Kernel 2: one wave per segment, 8 waves per block
    contrib_kernel<<<(B + 7) / 8, 256, 0, stream>>>(
        user_id, a_ids, a_seg, Ta, c_ids, c_seg, Tc, uf, ef, scores,
        pred, ca, cc, B);
}